// MRConv2d_85804856640065
// MI455X (gfx1250) — compile-verified
//
#include <hip/hip_runtime.h>

typedef __attribute__((ext_vector_type(2))) float v2f;
typedef __attribute__((ext_vector_type(8))) float v8f;

#define B_    4
#define C_    192
#define N_    3136
#define K_    16
#define COUT_ 384
#define NTILE 16             // n columns per workgroup (one WMMA N-tile)
#define NWG   (N_ / NTILE)   // 196 tiles per batch
#define MRS   17             // LDS mr row stride (bank-conflict padding)

// ---------------------------------------------------------------------------
// Shared phase 2: grouped 1x1 conv via V_WMMA_F32_16X16X4_F32 + bias + ReLU.
// wave w == conv group g.  A = weight (16x4), B = xin (4x16), C/D = f32 16x16.
// xin channels interleave as (x[c], mr[c]) pairs, so each lane's B fragment is
// one global x value (even row) + one LDS mr value (odd row).
// ---------------------------------------------------------------------------
__device__ __forceinline__ void phase2_wmma(
    const float* __restrict__ xb,      // x + b*C*N   ([C][N] layout)
    const float* __restrict__ weight,  // [Cout][96]
    const float* __restrict__ bias,    // [Cout]
    float*       __restrict__ outb,    // out + b*Cout*N
    const float* sMr,                  // [C][MRS] LDS
    int nBase, int tid)
{
    const int g    = tid >> 5;       // wave id == group id
    const int lane = tid & 31;
    const int col  = lane & 15;      // N column (n) / A row (m)
    const int hi   = lane >> 4;      // K half: K = 2*hi + {0,1}
    const int nG   = nBase + col;

    v8f acc[6];
    #pragma unroll
    for (int mt = 0; mt < 6; ++mt) {
        #pragma unroll
        for (int r = 0; r < 8; ++r)
            acc[mt][r] = bias[g * 96 + mt * 16 + r + 8 * hi];
    }

    for (int kk = 0; kk < 24; ++kk) {
        const int cch = g * 48 + kk * 2 + hi;       // x-channel for this K pair
        v2f bfrag;
        bfrag.x = xb[cch * N_ + nG];                // even interleaved ch: raw x
        bfrag.y = sMr[cch * MRS + col];             // odd  interleaved ch: mr
        #pragma unroll
        for (int mt = 0; mt < 6; ++mt) {
            const int co = g * 96 + mt * 16 + col;
            v2f afrag = *(const v2f*)(weight + co * 96 + kk * 4 + 2 * hi);
            acc[mt] = __builtin_amdgcn_wmma_f32_16x16x4_f32(
                false, afrag, false, bfrag, (short)0, acc[mt], false, false);
        }
    }

    #pragma unroll
    for (int mt = 0; mt < 6; ++mt) {
        #pragma unroll
        for (int r = 0; r < 8; ++r) {
            const int co = g * 96 + mt * 16 + r + 8 * hi;
            outb[co * N_ + nG] = fmaxf(acc[mt][r], 0.0f);
        }
    }
}

// ---------------------------------------------------------------------------
// Kernel 1: tiled transpose x[B][C][N] -> xt[B][N][C] (coalesced both sides)
// ---------------------------------------------------------------------------
__global__ __launch_bounds__(256) void transpose_xc(
    const float* __restrict__ x, float* __restrict__ xt)
{
    __shared__ float tile[32][33];
    int blk = blockIdx.x;
    const int nT = blk % (N_ / 32); blk /= (N_ / 32);
    const int cT = blk % (C_ / 32);
    const int b  = blk / (C_ / 32);
    const int tx = threadIdx.x & 31;
    const int ty = threadIdx.x >> 5;          // 8 rows
    const int n0 = nT * 32, c0 = cT * 32;
    const float* xb  = x  + b * (C_ * N_);
    float*       xtb = xt + b * (N_ * C_);

    #pragma unroll
    for (int i = 0; i < 32; i += 8)
        tile[ty + i][tx] = xb[(c0 + ty + i) * N_ + n0 + tx];
    __syncthreads();
    #pragma unroll
    for (int i = 0; i < 32; i += 8)
        xtb[(n0 + ty + i) * C_ + c0 + tx] = tile[tx][ty + i];
}

// ---------------------------------------------------------------------------
// Kernel 2 (fast path): mr via coalesced row loads from xt, then WMMA GEMM
// ---------------------------------------------------------------------------
__global__ __launch_bounds__(128) void mrconv2d_fused_xt(
    const float* __restrict__ x,      // [B, C, N]
    const int*   __restrict__ edge,   // [2, B, N, K]
    const float* __restrict__ weight, // [Cout, 96]
    const float* __restrict__ bias,   // [Cout]
    const float* __restrict__ xt,     // [B, N, C]
    float*       __restrict__ out)    // [B, Cout, N]
{
    __shared__ int   sJ[NTILE * K_];
    __shared__ int   sI[NTILE * K_];
    __shared__ float sMr[C_ * MRS];

    const int tid   = threadIdx.x;
    const int blk   = blockIdx.x;
    const int b     = blk / NWG;
    const int nBase = (blk % NWG) * NTILE;

    // Phase 0: stage edge indices
    {
        const int baseJ = b * (N_ * K_) + nBase * K_;          // edge_index[0]
        const int baseI = B_ * (N_ * K_) + baseJ;              // edge_index[1]
        for (int t = tid; t < NTILE * K_; t += 128) {
            sJ[t] = edge[baseJ + t];
            sI[t] = edge[baseI + t];
        }
    }
    __syncthreads();

    // Phase 1: mr rows from xt — fully coalesced b64 row loads.
    // Each lane owns channels {2*lane, 2*lane+1} + 64*p, p = 0..2.
    {
        const int w    = tid >> 5;
        const int lane = tid & 31;
        const float* xtb = xt + b * (N_ * C_);
        for (int nl = w; nl < NTILE; nl += 4) {
            v2f m[3];
            #pragma unroll
            for (int p = 0; p < 3; ++p) { m[p].x = -__builtin_inff(); m[p].y = -__builtin_inff(); }
            #pragma unroll 4
            for (int k = 0; k < K_; ++k) {
                const int j = sJ[nl * K_ + k];
                const int i = sI[nl * K_ + k];
                const float* rj = xtb + j * C_;
                const float* ri = xtb + i * C_;
                #pragma unroll
                for (int p = 0; p < 3; ++p) {
                    v2f vj = *(const v2f*)(rj + 2 * lane + 64 * p);
                    v2f vi = *(const v2f*)(ri + 2 * lane + 64 * p);
                    m[p].x = fmaxf(m[p].x, vj.x - vi.x);
                    m[p].y = fmaxf(m[p].y, vj.y - vi.y);
                }
            }
            #pragma unroll
            for (int p = 0; p < 3; ++p) {
                const int c = 2 * lane + 64 * p;
                sMr[(c + 0) * MRS + nl] = m[p].x;
                sMr[(c + 1) * MRS + nl] = m[p].y;
            }
        }
    }
    __syncthreads();

    phase2_wmma(x + b * (C_ * N_), weight, bias, out + b * (COUT_ * N_),
                sMr, nBase, tid);
}

// ---------------------------------------------------------------------------
// Kernel 2 (fallback, no workspace): mr via per-channel gathers from x
// ---------------------------------------------------------------------------
__global__ __launch_bounds__(128) void mrconv2d_fused_gather(
    const float* __restrict__ x,
    const int*   __restrict__ edge,
    const float* __restrict__ weight,
    const float* __restrict__ bias,
    float*       __restrict__ out)
{
    __shared__ int   sJ[NTILE * K_];
    __shared__ int   sI[NTILE * K_];
    __shared__ float sMr[C_ * MRS];

    const int tid   = threadIdx.x;
    const int blk   = blockIdx.x;
    const int b     = blk / NWG;
    const int nBase = (blk % NWG) * NTILE;

    {
        const int baseJ = b * (N_ * K_) + nBase * K_;
        const int baseI = B_ * (N_ * K_) + baseJ;
        for (int t = tid; t < NTILE * K_; t += 128) {
            sJ[t] = edge[baseJ + t];
            sI[t] = edge[baseI + t];
        }
    }
    __syncthreads();

    {
        const float* xb = x + b * (C_ * N_);
        const int nl   = tid & 15;
        const int cOff = tid >> 4;
        for (int c = cOff; c < C_; c += 8) {
            const float* row = xb + c * N_;
            float m = -__builtin_inff();
            #pragma unroll
            for (int k = 0; k < K_; ++k) {
                const int j = sJ[nl * K_ + k];
                const int i = sI[nl * K_ + k];
                m = fmaxf(m, row[j] - row[i]);
            }
            sMr[c * MRS + nl] = m;
        }
    }
    __syncthreads();

    phase2_wmma(x + b * (C_ * N_), weight, bias, out + b * (COUT_ * N_),
                sMr, nBase, tid);
}

extern "C" void kernel_launch(void* const* d_in, const int* in_sizes, int n_in,
                              void* d_out, int out_size, void* d_ws, size_t ws_size,
                              hipStream_t stream) {
    (void)in_sizes; (void)n_in; (void)out_size;
    const float* x    = (const float*)d_in[0];
    const int*   edge = (const int*)d_in[1];
    const float* wgt  = (const float*)d_in[2];
    const float* bias = (const float*)d_in[3];
    float*       out  = (float*)d_out;

    const size_t xtBytes = (size_t)B_ * N_ * C_ * sizeof(float);   // 9.6 MB

    if (ws_size >= xtBytes && d_ws != nullptr) {
        float* xt = (float*)d_ws;
        transpose_xc<<<dim3(B_ * (C_ / 32) * (N_ / 32)), dim3(256), 0, stream>>>(x, xt);
        mrconv2d_fused_xt<<<dim3(B_ * NWG), dim3(128), 0, stream>>>(
            x, edge, wgt, bias, xt, out);
    } else {
        mrconv2d_fused_gather<<<dim3(B_ * NWG), dim3(128), 0, stream>>>(
            x, edge, wgt, bias, out);
    }
}